// ENVRNetwork_53618371723965
// MI455X (gfx1250) — compile-verified
//
#include <hip/hip_runtime.h>
#include <math.h>

typedef float v2f __attribute__((ext_vector_type(2)));
typedef float v8f __attribute__((ext_vector_type(8)));

#define NPTS  (1 << 20)
#define NREG  512
#define GRID_ENT 35937   // 33^3
#define PI_F  3.14159274101257324f

__device__ __forceinline__ float lsq4(float v, float inv_s, float s) {
    return rintf(fminf(fmaxf(v * inv_s, -8.f), 7.f)) * s;
}
__device__ __forceinline__ float lsq8(float v, float s) {
    return rintf(fminf(fmaxf(v / s, -128.f), 127.f)) * s;
}

__global__ void k_zero(int* __restrict__ counts) {
    int i = blockIdx.x * blockDim.x + threadIdx.x;
    if (i < NREG) counts[i] = 0;
}

__global__ void k_count(const float* __restrict__ x, int* __restrict__ counts) {
    int p = blockIdx.x * blockDim.x + threadIdx.x;
    if (p >= NPTS) return;
    int rx = (int)floorf(8.f * x[(size_t)p * 7 + 0]);
    int ry = (int)floorf(8.f * x[(size_t)p * 7 + 1]);
    int rz = (int)floorf(8.f * x[(size_t)p * 7 + 2]);
    atomicAdd(&counts[64 * rz + 8 * ry + rx], 1);
}

__global__ void k_scan(const int* __restrict__ counts, int* __restrict__ starts,
                       int* __restrict__ cursor) {
    __shared__ int buf[NREG];
    int i = threadIdx.x;
    int c = counts[i];
    buf[i] = c;
    __syncthreads();
    for (int off = 1; off < NREG; off <<= 1) {
        int v = (i >= off) ? buf[i - off] : 0;
        __syncthreads();
        buf[i] += v;
        __syncthreads();
    }
    int excl = buf[i] - c;
    starts[i] = excl;
    cursor[i] = excl;
}

__global__ void k_scatter(const float* __restrict__ x, int* __restrict__ cursor,
                          int* __restrict__ sorted) {
    int p = blockIdx.x * blockDim.x + threadIdx.x;
    if (p >= NPTS) return;
    int rx = (int)floorf(8.f * x[(size_t)p * 7 + 0]);
    int ry = (int)floorf(8.f * x[(size_t)p * 7 + 1]);
    int rz = (int)floorf(8.f * x[(size_t)p * 7 + 2]);
    int r = 64 * rz + 8 * ry + rx;
    int pos = atomicAdd(&cursor[r], 1);
    sorted[pos] = p;
}

// One workgroup (8 waves) per region; each wave processes 16-point tiles with
// shared (region-uniform) weights via V_WMMA_F32_16X16X4_F32 chains.
__global__ __launch_bounds__(256) void k_main(
    const float* __restrict__ X, const float* __restrict__ grid,
    const float* __restrict__ grid_s,
    const float* __restrict__ w1g, const float* __restrict__ w1sp,
    const float* __restrict__ w2g, const float* __restrict__ w2sp,
    const float* __restrict__ w3g, const float* __restrict__ w3sp,
    const int* __restrict__ offset_table,
    const int* __restrict__ starts, const int* __restrict__ counts,
    const int* __restrict__ sorted, float* __restrict__ out)
{
    __shared__ float w1lds[16 * 16];   // K=16, N=16 (cols 8..15 zero)
    __shared__ float w2lds[28 * 32];   // K=28 (rows 26,27 zero), N=32
    __shared__ float w3lds[32 * 16];   // K=32, N=16 (cols 3..15 zero)
    __shared__ float hbuf1[8][16 * 16]; // layer1 input  [m][k]
    __shared__ float hbuf [8][16 * 28]; // layer2 input  [m][k] = dens(8)|sh(18)|0,0
    __shared__ float h2buf[8][16 * 32]; // layer3 input  [m][k]
    __shared__ int   idxbuf[8][16];

    const int tid = threadIdx.x;
    const int ri  = blockIdx.x;
    const int o   = offset_table[ri];
    const float s1 = w1sp[0], s2 = w2sp[0], s3 = w3sp[0];
    const float sg = grid_s[0];
    const float inv_sg = 1.f / sg;

    // --- stage dequantized weights into LDS ---
    {
        int k = tid >> 4, n = tid & 15;
        w1lds[tid] = (n < 8) ? lsq8(w1g[((size_t)o * 16 + k) * 8 + n], s1) : 0.f;
    }
    for (int j = tid; j < 28 * 32; j += 256) {
        int k = j >> 5, n = j & 31;
        w2lds[j] = (k < 26) ? lsq8(w2g[((size_t)o * 26 + k) * 32 + n], s2) : 0.f;
    }
    for (int j = tid; j < 32 * 16; j += 256) {
        int k = j >> 4, n = j & 15;
        w3lds[j] = (n < 3) ? lsq8(w3g[((size_t)o * 32 + k) * 3 + n], s3) : 0.f;
    }
    __syncthreads();

    const int cnt  = counts[ri];
    const int base = starts[ri];
    const int wv   = tid >> 5;
    const int lane = tid & 31;
    const int m    = lane & 15;     // point-in-tile / N column
    const int half = lane >> 4;     // channel / K half

    for (int t = wv; t * 16 < cnt; t += 8) {
        const int tb = t * 16;
        const int nvalid = min(16, cnt - tb);
        const int mm = min(m, nvalid - 1);
        const int p  = sorted[base + tb + mm];
        if (half == 0) idxbuf[wv][m] = p;

        // --- trilinear setup (both halves redundantly) ---
        float px = 8.f * X[(size_t)p * 7 + 0];
        float py = 8.f * X[(size_t)p * 7 + 1];
        float pz = 8.f * X[(size_t)p * 7 + 2];
        float gx = 32.f * (px - floorf(px));
        float gy = 32.f * (py - floorf(py));
        float gz = 32.f * (pz - floorf(pz));
        float fgx = floorf(gx), fgy = floorf(gy), fgz = floorf(gz);
        int pgx = min((int)fgx, 31), pgy = min((int)fgy, 31), pgz = min((int)fgz, 31);
        float frx = gx - fgx, fry = gy - fgy, frz = gz - fgz;

        // --- gather 8 corners; this lane owns channels [half*8, half*8+8) ---
        float acc[8] = {0.f, 0.f, 0.f, 0.f, 0.f, 0.f, 0.f, 0.f};
        const float* gslab = grid + (size_t)o * (GRID_ENT * 16) + half * 8;
        #pragma unroll
        for (int c = 0; c < 8; ++c) {
            int ic = (pgx + ((c >> 2) & 1)) + (pgy + ((c >> 1) & 1)) * 33 +
                     (pgz + (c & 1)) * 1089;
            float wc = ((c & 4) ? frx : 1.f - frx) *
                       ((c & 2) ? fry : 1.f - fry) *
                       ((c & 1) ? frz : 1.f - frz);
            const float4* src = (const float4*)(gslab + (size_t)ic * 16);
            float4 a = src[0], b = src[1];
            acc[0] = fmaf(wc, lsq4(a.x, inv_sg, sg), acc[0]);
            acc[1] = fmaf(wc, lsq4(a.y, inv_sg, sg), acc[1]);
            acc[2] = fmaf(wc, lsq4(a.z, inv_sg, sg), acc[2]);
            acc[3] = fmaf(wc, lsq4(a.w, inv_sg, sg), acc[3]);
            acc[4] = fmaf(wc, lsq4(b.x, inv_sg, sg), acc[4]);
            acc[5] = fmaf(wc, lsq4(b.y, inv_sg, sg), acc[5]);
            acc[6] = fmaf(wc, lsq4(b.z, inv_sg, sg), acc[6]);
            acc[7] = fmaf(wc, lsq4(b.w, inv_sg, sg), acc[7]);
        }
        #pragma unroll
        for (int i = 0; i < 8; ++i)
            hbuf1[wv][m * 16 + half * 8 + i] = fmaxf(acc[i], 0.f);  // relu(pos_enc)

        // --- sh encoding: sh[comp*6 + fq*2 + {0:sin,1:cos}] at cols 8..25 ---
        if (half == 0) {
            #pragma unroll
            for (int comp = 0; comp < 2; ++comp) {
                float d = X[(size_t)p * 7 + 4 + comp] * 2.f - 1.f;
                #pragma unroll
                for (int fq = 0; fq < 3; ++fq) {
                    float a = PI_F * d * (float)(1 << fq);
                    hbuf[wv][m * 28 + 8 + comp * 6 + fq * 2 + 0] = sinf(a);
                    hbuf[wv][m * 28 + 8 + comp * 6 + fq * 2 + 1] = cosf(a);
                }
            }
        } else {
            float d = X[(size_t)p * 7 + 6] * 2.f - 1.f;
            #pragma unroll
            for (int fq = 0; fq < 3; ++fq) {
                float a = PI_F * d * (float)(1 << fq);
                hbuf[wv][m * 28 + 8 + 12 + fq * 2 + 0] = sinf(a);
                hbuf[wv][m * 28 + 8 + 12 + fq * 2 + 1] = cosf(a);
            }
            hbuf[wv][m * 28 + 26] = 0.f;
            hbuf[wv][m * 28 + 27] = 0.f;
        }

        // --- layer 1: 16(M) x 16(K) x 16(N, 8 valid) : density ---
        v8f c1 = {0.f, 0.f, 0.f, 0.f, 0.f, 0.f, 0.f, 0.f};
        #pragma unroll
        for (int kk = 0; kk < 4; ++kk) {
            int k0 = kk * 4 + half * 2;
            v2f A; A.x = hbuf1[wv][m * 16 + k0]; A.y = hbuf1[wv][m * 16 + k0 + 1];
            v2f B; B.x = w1lds[k0 * 16 + m];     B.y = w1lds[(k0 + 1) * 16 + m];
            c1 = __builtin_amdgcn_wmma_f32_16x16x4_f32(false, A, false, B,
                                                       (short)0, c1, false, false);
        }
        if (m < 8) {   // density (no relu) -> hbuf cols 0..7
            #pragma unroll
            for (int r = 0; r < 8; ++r)
                hbuf[wv][(r + half * 8) * 28 + m] = c1[r];
        }

        // --- layer 2: 16 x 28(K) x 32(N) ---
        #pragma unroll
        for (int nt = 0; nt < 2; ++nt) {
            v8f c2 = {0.f, 0.f, 0.f, 0.f, 0.f, 0.f, 0.f, 0.f};
            #pragma unroll
            for (int kk = 0; kk < 7; ++kk) {
                int k0 = kk * 4 + half * 2;
                v2f A; A.x = hbuf[wv][m * 28 + k0]; A.y = hbuf[wv][m * 28 + k0 + 1];
                v2f B; B.x = w2lds[k0 * 32 + nt * 16 + m];
                       B.y = w2lds[(k0 + 1) * 32 + nt * 16 + m];
                c2 = __builtin_amdgcn_wmma_f32_16x16x4_f32(false, A, false, B,
                                                           (short)0, c2, false, false);
            }
            #pragma unroll
            for (int r = 0; r < 8; ++r)
                h2buf[wv][(r + half * 8) * 32 + nt * 16 + m] = fmaxf(c2[r], 0.f);
        }

        // --- layer 3: 16 x 32(K) x 16(N, 3 valid) : rgb ---
        v8f c3 = {0.f, 0.f, 0.f, 0.f, 0.f, 0.f, 0.f, 0.f};
        #pragma unroll
        for (int kk = 0; kk < 8; ++kk) {
            int k0 = kk * 4 + half * 2;
            v2f A; A.x = h2buf[wv][m * 32 + k0]; A.y = h2buf[wv][m * 32 + k0 + 1];
            v2f B; B.x = w3lds[k0 * 16 + m];     B.y = w3lds[(k0 + 1) * 16 + m];
            c3 = __builtin_amdgcn_wmma_f32_16x16x4_f32(false, A, false, B,
                                                       (short)0, c3, false, false);
        }

        // --- scatter out[p] = (rgb0, rgb1, rgb2, density0) ---
        #pragma unroll
        for (int r = 0; r < 8; ++r) {
            int M = r + half * 8;
            if (M < nvalid) {
                int pi = idxbuf[wv][M];
                if (m < 3)  out[(size_t)pi * 4 + m] = c3[r];
                if (m == 0) out[(size_t)pi * 4 + 3] = c1[r];
            }
        }
    }
}

extern "C" void kernel_launch(void* const* d_in, const int* in_sizes, int n_in,
                              void* d_out, int out_size, void* d_ws, size_t ws_size,
                              hipStream_t stream) {
    const float* X    = (const float*)d_in[0];
    const float* grid = (const float*)d_in[1];
    const float* gls  = (const float*)d_in[2];
    const float* w1   = (const float*)d_in[3];
    const float* w1s  = (const float*)d_in[4];
    const float* w2   = (const float*)d_in[5];
    const float* w2s  = (const float*)d_in[6];
    const float* w3   = (const float*)d_in[7];
    const float* w3s  = (const float*)d_in[8];
    const int*   otab = (const int*)d_in[9];
    float* out = (float*)d_out;

    int* counts = (int*)d_ws;
    int* starts = counts + NREG;
    int* cursor = starts + NREG;
    int* sorted = cursor + NREG;

    hipLaunchKernelGGL(k_zero, dim3(2), dim3(256), 0, stream, counts);
    hipLaunchKernelGGL(k_count, dim3(NPTS / 256), dim3(256), 0, stream, X, counts);
    hipLaunchKernelGGL(k_scan, dim3(1), dim3(NREG), 0, stream, counts, starts, cursor);
    hipLaunchKernelGGL(k_scatter, dim3(NPTS / 256), dim3(256), 0, stream, X, cursor, sorted);
    hipLaunchKernelGGL(k_main, dim3(NREG), dim3(256), 0, stream,
                       X, grid, gls, w1, w1s, w2, w2s, w3, w3s, otab,
                       starts, counts, sorted, out);
}